// Embedding_77541339562063
// MI455X (gfx1250) — compile-verified
//
#include <hip/hip_runtime.h>
#include <cstdint>

#define VOCAB 32000
#define EMBED 128

// ---- gfx1250 async LDS-DMA availability probe (compile-safe) ----------------
#if __has_builtin(__builtin_amdgcn_global_load_async_to_lds_b32) && \
    __has_builtin(__builtin_amdgcn_global_store_async_from_lds_b128)
#define HAVE_ASYNC_LDS 1
#else
#define HAVE_ASYNC_LDS 0
#endif

typedef int v4i __attribute__((ext_vector_type(4)));

typedef __attribute__((address_space(1))) int  as1_int;
typedef __attribute__((address_space(3))) int  as3_int;
typedef __attribute__((address_space(1))) v4i  as1_v4i;
typedef __attribute__((address_space(3))) v4i  as3_v4i;

__device__ __forceinline__ void wait_asynccnt0() {
#if __has_builtin(__builtin_amdgcn_s_wait_asynccnt)
  __builtin_amdgcn_s_wait_asynccnt(0);
#else
  asm volatile("s_wait_asynccnt 0" ::: "memory");
#endif
}

// Global (generic) pointer -> addrspace(1) typed pointer (identity bits).
__device__ __forceinline__ as1_int*  g32(const float* p) {
  return (as1_int*)(unsigned long long)(uintptr_t)p;
}
__device__ __forceinline__ as1_v4i*  g128(float* p) {
  return (as1_v4i*)(unsigned long long)(uintptr_t)p;
}
// Generic LDS pointer -> addrspace(3): low 32 bits are the LDS byte offset
// (ISA §10.2 aperture rules); compiler folds the cast chain for __shared__.
__device__ __forceinline__ as3_int*  l32(const float* p) {
  return (as3_int*)(uint32_t)(uintptr_t)p;
}
__device__ __forceinline__ as3_v4i*  l128(const float* p) {
  return (as3_v4i*)(uint32_t)(uintptr_t)p;
}

// One wave (32 lanes) gathers one token's 128-float embedding row.
// Lane l owns embed elements 4l..4l+3 (stride-VOCAB column loads),
// output is one coalesced float4 per lane (512B contiguous per wave).
__global__ __launch_bounds__(256) void emb_gather_kernel(
    const int* __restrict__ X, const float* __restrict__ W,
    float* __restrict__ out, int ntok) {
  const int lane = threadIdx.x & 31;
  const int wib  = threadIdx.x >> 5;          // wave index in block
  const int wpb  = blockDim.x >> 5;           // waves per block (8)
  const int wave = blockIdx.x * wpb + wib;
  const int nw   = gridDim.x * wpb;           // total waves

#if HAVE_ASYNC_LDS
  __shared__ float tile[8][EMBED];            // 512 B per wave, private per wib
#endif

  for (int t = wave; t < ntok; t += nw) {
    const int v = X[t];                       // wave-uniform index

    // Prefetch next iteration's strided column (gfx1250 global_prefetch_b8).
    const int tn = t + nw;
    if (tn < ntok) {
      const int vn = X[tn];
      const float* pf = W + (size_t)(4 * lane) * VOCAB + vn;
      __builtin_prefetch(pf,             0, 1);
      __builtin_prefetch(pf +     VOCAB, 0, 1);
      __builtin_prefetch(pf + 2 * VOCAB, 0, 1);
      __builtin_prefetch(pf + 3 * VOCAB, 0, 1);
    }

    const float* col = W + (size_t)(4 * lane) * VOCAB + v;
    float* dst = out + (size_t)t * EMBED + 4 * lane;

#if HAVE_ASYNC_LDS
    // Pure DMA path: gather L2 -> LDS (per-lane addresses), then burst
    // LDS -> global as b128. Data never occupies VGPRs. Tracked on ASYNCcnt.
    float* lp = &tile[wib][4 * lane];
    __builtin_amdgcn_global_load_async_to_lds_b32(g32(col),             l32(lp + 0), 0, 0);
    __builtin_amdgcn_global_load_async_to_lds_b32(g32(col +     VOCAB), l32(lp + 1), 0, 0);
    __builtin_amdgcn_global_load_async_to_lds_b32(g32(col + 2 * VOCAB), l32(lp + 2), 0, 0);
    __builtin_amdgcn_global_load_async_to_lds_b32(g32(col + 3 * VOCAB), l32(lp + 3), 0, 0);
    wait_asynccnt0();                         // loads landed in LDS
    __builtin_amdgcn_global_store_async_from_lds_b128(g128(dst), l128(lp), 0, 0);
    wait_asynccnt0();                         // LDS reusable next iteration
#else
    // VGPR gather fallback: 4 strided b32 loads, one coalesced b128 store.
    float4 r;
    r.x = col[0];
    r.y = col[VOCAB];
    r.z = col[2 * VOCAB];
    r.w = col[3 * VOCAB];
    *reinterpret_cast<float4*>(dst) = r;
#endif
  }
}

extern "C" void kernel_launch(void* const* d_in, const int* in_sizes, int n_in,
                              void* d_out, int out_size, void* d_ws, size_t ws_size,
                              hipStream_t stream) {
  const int*   X   = (const int*)d_in[0];     // [2, 2048] int32 token ids
  const float* W   = (const float*)d_in[1];   // [128, 32000] float32
  float*       out = (float*)d_out;           // [2, 2048, 128] float32
  const int ntok = in_sizes[0];               // 4096 tokens

  // 256 blocks x 8 waves = 2048 waves -> 2 tokens per wave (prefetch useful),
  // full waves, no divergence (4096 % 2048 == 0).
  dim3 grid(256), block(256);
  emb_gather_kernel<<<grid, block, 0, stream>>>(X, W, out, ntok);
}